// MetaNet_Gated_48945447306073
// MI455X (gfx1250) — compile-verified
//
#include <hip/hip_runtime.h>

typedef float v2f __attribute__((ext_vector_type(2)));
typedef float v4f __attribute__((ext_vector_type(4)));
typedef float v8f __attribute__((ext_vector_type(8)));

#define BATCH 256
#define VIS   1024
#define BOT   64
#define CTX   512
#define HDIM  16
#define MLPH  4
#define NSPLIT 4              // split-K factor over the KAN i-reduction

// ---------------------------------------------------------------------------
// K1: z = x @ reduce_w + reduce_b   (256,1024)@(1024,64) via fp32 WMMA 16x16x4
// ---------------------------------------------------------------------------
__global__ __launch_bounds__(256) void k_reduce_gemm(
    const float* __restrict__ x, const float* __restrict__ w,
    const float* __restrict__ bias, float* __restrict__ z)
{
    const int lane = threadIdx.x & 31;
    const int wave = threadIdx.x >> 5;
    const int tile = blockIdx.x * 8 + wave;   // 0..63
    const int bt   = tile >> 2;
    const int nt   = tile & 3;
    const int ml   = lane & 15;
    const int hi   = lane >> 4;
    const int row  = bt * 16 + ml;
    const int col  = nt * 16 + ml;

    v8f c = {};
    for (int k = 0; k < VIS; k += 4) {
        const int kb = k + 2 * hi;
        v2f a;
        a.x = x[row * VIS + kb];
        a.y = x[row * VIS + kb + 1];
        v2f b;
        b.x = w[kb * BOT + col];
        b.y = w[(kb + 1) * BOT + col];
        c = __builtin_amdgcn_wmma_f32_16x16x4_f32(
                false, a, false, b, (short)0, c, false, false);
    }
#pragma unroll
    for (int r = 0; r < 8; ++r) {
        const int mm = bt * 16 + r + 8 * hi;
        const int nn = nt * 16 + ml;
        z[mm * BOT + nn] = c[r] + bias[nn];
    }
}

// ---------------------------------------------------------------------------
// KAN inner accumulation over i in [i0,i1): explicit b128 weight loads.
// zcol(r) supplies z[b-row r][i] (broadcast from LDS by the caller lambda-free
// convention: caller passes a pointer to its LDS sub-tile and the i stride).
// ---------------------------------------------------------------------------
__device__ __forceinline__ void kan_accum(
    const float* __restrict__ zsub, int zstride, int mbase,
    const float* __restrict__ kan_W1, const float* __restrict__ kan_b1,
    const float* __restrict__ kan_W2, const float* __restrict__ kan_b2,
    int o, int i0, int i1, float acc[8], float& kb2)
{
    for (int i = i0; i < i1; ++i) {
        const size_t base = ((size_t)i * CTX + o) * HDIM;
        const v4f* __restrict__ w1p = (const v4f*)(kan_W1 + base);
        const v4f* __restrict__ b1p = (const v4f*)(kan_b1 + base);
        const v4f* __restrict__ w2p = (const v4f*)(kan_W2 + base);
        if (i + 1 < i1) {
            const size_t nb = ((size_t)(i + 1) * CTX + o) * HDIM;
            __builtin_prefetch(kan_W1 + nb, 0, 0);
            __builtin_prefetch(kan_W2 + nb, 0, 0);
        }
        kb2 += kan_b2[i * CTX + o];

        float zi[8];
#pragma unroll
        for (int r = 0; r < 8; ++r) zi[r] = zsub[(mbase + r) * zstride + (i - i0)];

#pragma unroll
        for (int q = 0; q < HDIM / 4; ++q) {
            const v4f w1 = w1p[q];
            const v4f b1 = b1p[q];
            const v4f w2 = w2p[q];
#pragma unroll
            for (int c = 0; c < 4; ++c) {
#pragma unroll
                for (int r = 0; r < 8; ++r) {
                    float t = fmaf(zi[r], w1[c], b1[c]);
                    t = fmaxf(t, 0.0f);
                    acc[r] = fmaf(t, w2[c], acc[r]);
                }
            }
        }
    }
}

// ---------------------------------------------------------------------------
// K2a (split-K path): partial KAN sums over 16 i's per block.z.
// grid (16 btile, 8 otile, NSPLIT), 128 threads. partial[s][b][o] in ws.
// ---------------------------------------------------------------------------
__global__ __launch_bounds__(128) void k_kan_partial(
    const float* __restrict__ z,
    const float* __restrict__ kan_W1, const float* __restrict__ kan_b1,
    const float* __restrict__ kan_W2, const float* __restrict__ kan_b2,
    float* __restrict__ partial)
{
    __shared__ float zsub[16][17];     // 16 rows x 16 i-columns (+pad)
    const int tid  = threadIdx.x;
    const int lane = tid & 31;
    const int wave = tid >> 5;
    const int b0   = blockIdx.x * 16;
    const int o0   = blockIdx.y * 64;
    const int s    = blockIdx.z;
    const int i0   = s * (BOT / NSPLIT);
    const int i1   = i0 + (BOT / NSPLIT);

    for (int idx = tid; idx < 16 * 16; idx += 128) {
        const int bb = idx >> 4, ii = idx & 15;
        zsub[bb][ii] = z[(b0 + bb) * BOT + i0 + ii];
    }
    __syncthreads();

    const int o     = o0 + wave * 16 + (lane & 15);
    const int mbase = 8 * (lane >> 4);

    float acc[8];
#pragma unroll
    for (int r = 0; r < 8; ++r) acc[r] = 0.0f;
    float kb2 = 0.0f;

    kan_accum(&zsub[0][0], 17, mbase, kan_W1, kan_b1, kan_W2, kan_b2,
              o, i0, i1, acc, kb2);

#pragma unroll
    for (int r = 0; r < 8; ++r) {
        const int mm = b0 + mbase + r;
        partial[((size_t)s * BATCH + mm) * CTX + o] = acc[r] + kb2;
    }
}

// ---------------------------------------------------------------------------
// K2b (split-K path): gate WMMA + ann WMMA + partial reduction + blend.
// grid (16, 8), 128 threads.
// ---------------------------------------------------------------------------
__global__ __launch_bounds__(128) void k_combine(
    const float* __restrict__ z,
    const float* __restrict__ ann_w1, const float* __restrict__ ann_b1,
    const float* __restrict__ ann_w2, const float* __restrict__ ann_b2,
    const float* __restrict__ gate_w, const float* __restrict__ gate_b,
    const float* __restrict__ partial, float* __restrict__ out)
{
    __shared__ float zs[16][66];
    __shared__ float hid[16][MLPH];
    const int tid  = threadIdx.x;
    const int lane = tid & 31;
    const int wave = tid >> 5;
    const int b0   = blockIdx.x * 16;
    const int o0   = blockIdx.y * 64;

    for (int idx = tid; idx < 16 * BOT; idx += 128) {
        const int bb = idx >> 6, ii = idx & 63;
        zs[bb][ii] = z[(b0 + bb) * BOT + ii];
    }
    __syncthreads();

    if (tid < 16 * MLPH) {
        const int bb = tid >> 2, mm = tid & 3;
        float sum = ann_b1[mm];
        for (int i = 0; i < BOT; ++i)
            sum = fmaf(zs[bb][i], ann_w1[i * MLPH + mm], sum);
        hid[bb][mm] = fmaxf(sum, 0.0f);
    }
    __syncthreads();

    const int ml = lane & 15;
    const int hi = lane >> 4;
    const int o  = o0 + wave * 16 + ml;

    v8f cg = {};
    for (int k = 0; k < BOT; k += 4) {
        const int kb = k + 2 * hi;
        v2f a; a.x = zs[ml][kb] + 1.0f; a.y = zs[ml][kb + 1] + 1.0f;
        v2f b; b.x = gate_w[kb * CTX + o]; b.y = gate_w[(kb + 1) * CTX + o];
        cg = __builtin_amdgcn_wmma_f32_16x16x4_f32(
                 false, a, false, b, (short)0, cg, false, false);
    }
    v8f ca = {};
    {
        const int kb = 2 * hi;
        v2f a; a.x = hid[ml][kb]; a.y = hid[ml][kb + 1];
        v2f b; b.x = ann_w2[kb * CTX + o]; b.y = ann_w2[(kb + 1) * CTX + o];
        ca = __builtin_amdgcn_wmma_f32_16x16x4_f32(
                 false, a, false, b, (short)0, ca, false, false);
    }

    const int   mbase = 8 * hi;
    const float gb = gate_b[o];
    const float ab = ann_b2[o];
#pragma unroll
    for (int r = 0; r < 8; ++r) {
        const int mm = b0 + mbase + r;
        float kanv = 0.0f;
#pragma unroll
        for (int s = 0; s < NSPLIT; ++s)
            kanv += partial[((size_t)s * BATCH + mm) * CTX + o];
        const float g = 1.0f / (1.0f + __expf(-(cg[r] + gb)));
        out[(size_t)mm * CTX + o] = g * (ca[r] + ab) + (1.0f - g) * kanv;
    }
}

// ---------------------------------------------------------------------------
// Fallback fused kernel (small-workspace path): same as round 1 but with
// b128 weight loads. grid (16, 8), 128 threads.
// ---------------------------------------------------------------------------
__global__ __launch_bounds__(128) void k_fused(
    const float* __restrict__ z,
    const float* __restrict__ ann_w1, const float* __restrict__ ann_b1,
    const float* __restrict__ ann_w2, const float* __restrict__ ann_b2,
    const float* __restrict__ kan_W1, const float* __restrict__ kan_b1,
    const float* __restrict__ kan_W2, const float* __restrict__ kan_b2,
    const float* __restrict__ gate_w, const float* __restrict__ gate_b,
    float* __restrict__ out)
{
    __shared__ float zs[16][66];
    __shared__ float hid[16][MLPH];
    const int tid  = threadIdx.x;
    const int lane = tid & 31;
    const int wave = tid >> 5;
    const int b0   = blockIdx.x * 16;
    const int o0   = blockIdx.y * 64;

    for (int idx = tid; idx < 16 * BOT; idx += 128) {
        const int bb = idx >> 6, ii = idx & 63;
        zs[bb][ii] = z[(b0 + bb) * BOT + ii];
    }
    __syncthreads();

    if (tid < 16 * MLPH) {
        const int bb = tid >> 2, mm = tid & 3;
        float sum = ann_b1[mm];
        for (int i = 0; i < BOT; ++i)
            sum = fmaf(zs[bb][i], ann_w1[i * MLPH + mm], sum);
        hid[bb][mm] = fmaxf(sum, 0.0f);
    }
    __syncthreads();

    const int ml = lane & 15;
    const int hi = lane >> 4;
    const int o  = o0 + wave * 16 + ml;

    v8f cg = {};
    for (int k = 0; k < BOT; k += 4) {
        const int kb = k + 2 * hi;
        v2f a; a.x = zs[ml][kb] + 1.0f; a.y = zs[ml][kb + 1] + 1.0f;
        v2f b; b.x = gate_w[kb * CTX + o]; b.y = gate_w[(kb + 1) * CTX + o];
        cg = __builtin_amdgcn_wmma_f32_16x16x4_f32(
                 false, a, false, b, (short)0, cg, false, false);
    }
    v8f ca = {};
    {
        const int kb = 2 * hi;
        v2f a; a.x = hid[ml][kb]; a.y = hid[ml][kb + 1];
        v2f b; b.x = ann_w2[kb * CTX + o]; b.y = ann_w2[(kb + 1) * CTX + o];
        ca = __builtin_amdgcn_wmma_f32_16x16x4_f32(
                 false, a, false, b, (short)0, ca, false, false);
    }

    const int mbase = 8 * hi;
    float acc[8];
#pragma unroll
    for (int r = 0; r < 8; ++r) acc[r] = 0.0f;
    float kb2 = 0.0f;

    kan_accum(&zs[0][0], 66, mbase, kan_W1, kan_b1, kan_W2, kan_b2,
              o, 0, BOT, acc, kb2);

    const float gb = gate_b[o];
    const float ab = ann_b2[o];
#pragma unroll
    for (int r = 0; r < 8; ++r) {
        const float g = 1.0f / (1.0f + __expf(-(cg[r] + gb)));
        const int  mm = b0 + mbase + r;
        out[(size_t)mm * CTX + o] = g * (ca[r] + ab) + (1.0f - g) * (acc[r] + kb2);
    }
}

extern "C" void kernel_launch(void* const* d_in, const int* in_sizes, int n_in,
                              void* d_out, int out_size, void* d_ws, size_t ws_size,
                              hipStream_t stream) {
    const float* x        = (const float*)d_in[0];
    const float* reduce_w = (const float*)d_in[1];
    const float* reduce_b = (const float*)d_in[2];
    const float* ann_w1   = (const float*)d_in[3];
    const float* ann_b1   = (const float*)d_in[4];
    const float* ann_w2   = (const float*)d_in[5];
    const float* ann_b2   = (const float*)d_in[6];
    const float* kan_W1   = (const float*)d_in[7];
    const float* kan_b1   = (const float*)d_in[8];
    const float* kan_W2   = (const float*)d_in[9];
    const float* kan_b2   = (const float*)d_in[10];
    const float* gate_w   = (const float*)d_in[11];
    const float* gate_b   = (const float*)d_in[12];

    float* z   = (float*)d_ws;                      // 64 KB
    float* out = (float*)d_out;

    const size_t zbytes = (size_t)BATCH * BOT * sizeof(float);          // 64 KB
    const size_t pbytes = (size_t)NSPLIT * BATCH * CTX * sizeof(float); // 2 MB

    k_reduce_gemm<<<8, 256, 0, stream>>>(x, reduce_w, reduce_b, z);

    if (ws_size >= zbytes + pbytes) {
        // high-occupancy split-K path: 2048 waves on the KAN reduction
        float* partial = (float*)((char*)d_ws + zbytes);
        dim3 gp(16, 8, NSPLIT);
        k_kan_partial<<<gp, 128, 0, stream>>>(z, kan_W1, kan_b1, kan_W2, kan_b2,
                                              partial);
        dim3 gc(16, 8);
        k_combine<<<gc, 128, 0, stream>>>(z, ann_w1, ann_b1, ann_w2, ann_b2,
                                          gate_w, gate_b, partial, out);
    } else {
        dim3 grid(16, 8);
        k_fused<<<grid, 128, 0, stream>>>(z, ann_w1, ann_b1, ann_w2, ann_b2,
                                          kan_W1, kan_b1, kan_W2, kan_b2,
                                          gate_w, gate_b, out);
    }
}